// DFTQNN_29661044146723
// MI455X (gfx1250) — compile-verified
//
#include <hip/hip_runtime.h>

#define NW      21
#define DIM_Q   (1u << NW)          // 2,097,152 amplitudes (32MB complex128)
#define NGATES  32
#define NMEAS   8

// Fused-group kernel geometry
#define EB      12                  // log2 amplitudes per block
#define BAMPS   (1u << EB)          // 4096 amplitudes = 64KB LDS per block
#define TPB     1024
#define EPT     (BAMPS / TPB)       // 4 elements / thread
#define PPT     (BAMPS / 2 / TPB)   // 2 pairs / thread
#define RMAX    (EB - 3)            // 9: keep >=8 consecutive reps per block (cacheline locality)
#define NGMAX   4                   // each closed group absorbs >= RMAX gates => <= ceil(32/9)=4 groups

// ---- CDNA5 async global<->LDS paths (guarded; fall back to plain ld/ds) ----
#if __has_builtin(__builtin_amdgcn_global_load_async_to_lds_b128)
#define QS_ASYNC_LD 1
#endif
#if __has_builtin(__builtin_amdgcn_global_store_async_from_lds_b128)
#define QS_ASYNC_ST 1
#endif
#if __has_builtin(__builtin_amdgcn_s_wait_asynccnt)
#define QS_WAIT_ASYNC() __builtin_amdgcn_s_wait_asynccnt(0)
#else
#define QS_WAIT_ASYNC() asm volatile("s_wait_asynccnt 0" ::: "memory")
#endif

#if defined(QS_ASYNC_LD) || defined(QS_ASYNC_ST)
// b128 payload type expected by the builtins: int __vector_size__(16)
typedef int qs_v4i __attribute__((vector_size(4 * sizeof(int))));
typedef __attribute__((address_space(1))) qs_v4i* qs_gptr;   // global
typedef __attribute__((address_space(3))) qs_v4i* qs_lptr;   // LDS
#endif

// Device-side precomputed gate grouping (host cannot read device gate masks
// under graph capture, so the GF(2) echelon runs on device).
struct QPre {
    int      ngroups;
    int      gstart[NGMAX], gend[NGMAX], grank[NGMAX];
    unsigned pivmask[NGMAX];
    unsigned basis[NGMAX][RMAX];    // reduced-echelon basis, MSB pivots
    unsigned npbit[NGMAX][NW];      // non-pivot bit positions, ascending
    unsigned phi[NGATES];           // flip in local (basis) coordinates
    unsigned pmloc[NGATES];         // bit i = parity(basis[i] & pmask)
    unsigned upml[NGATES];          // bit k = (pmask >> npbit[k]) & 1
};

// ---------------------------------------------------------------------------
// Kernel 1: per-gate cos/sin + zero the 9 fp64 accumulators.
// ---------------------------------------------------------------------------
__global__ void qsim_init(const float* __restrict__ theta,
                          double* __restrict__ cs,
                          double* __restrict__ acc) {
    int t = threadIdx.x;
    if (t < NGATES) {
        double th = 0.5 * (double)theta[t];
        cs[2 * t + 0] = cos(th);
        cs[2 * t + 1] = sin(th);
    }
    if (t < NMEAS + 1) acc[t] = 0.0;
}

// ---------------------------------------------------------------------------
// Kernel 2: GF(2) echelon precompute (single thread; ~1k scalar ops).
// Groups consecutive gates while rank(span of flips) <= RMAX; closing a group
// requires rank==RMAX which requires >= RMAX absorbed gates => ngroups <= 4.
// Pass 1 fixes boundaries + final reduced basis; pass 2 derives per-gate
// local coordinates against the *final* basis.
// ---------------------------------------------------------------------------
__global__ void qsim_pre(const int* __restrict__ gflip,
                         const int* __restrict__ gpmask,
                         QPre* __restrict__ pre) {
    if (threadIdx.x != 0 || blockIdx.x != 0) return;
    unsigned bas[RMAX]; int piv[RMAX];
    int ng = 0, gate = 0;
    while (gate < NGATES && ng < NGMAX) {
        int r = 0;
        int start = gate;
        while (gate < NGATES) {
            unsigned v = (unsigned)gflip[gate];
            for (int i = 0; i < r; ++i)
                if ((v >> piv[i]) & 1u) v ^= bas[i];
            if (v == 0u) { gate++; continue; }          // in span (incl. all-Z)
            if (r == RMAX) break;                       // close group
            int p = 31 - __clz((int)v);                 // MSB pivot -> low bits stay non-pivot
            for (int i = 0; i < r; ++i)                 // keep reduced form
                if ((bas[i] >> p) & 1u) bas[i] ^= v;
            bas[r] = v; piv[r] = p; r++;
            gate++;
        }
        pre->gstart[ng] = start; pre->gend[ng] = gate; pre->grank[ng] = r;
        unsigned pm = 0;
        for (int i = 0; i < r; ++i) { pm |= 1u << piv[i]; pre->basis[ng][i] = bas[i]; }
        pre->pivmask[ng] = pm;
        int k = 0;
        for (int b = 0; b < NW; ++b)
            if (!((pm >> b) & 1u)) pre->npbit[ng][k++] = (unsigned)b;
        for (int g = start; g < gate; ++g) {
            unsigned f = (unsigned)gflip[g], pmask = (unsigned)gpmask[g];
            unsigned phi = 0, pml = 0, upml = 0;
            for (int i = 0; i < r; ++i) {
                phi |= ((f >> piv[i]) & 1u) << i;       // reduced form: coeff = pivot bit
                pml |= (unsigned)(__popc(bas[i] & pmask) & 1) << i;
            }
            for (int b = 0; b < NW - r; ++b)
                upml |= ((pmask >> pre->npbit[ng][b]) & 1u) << b;
            pre->phi[g] = phi; pre->pmloc[g] = pml; pre->upml[g] = upml;
        }
        ng++;
    }
    pre->ngroups = ng;
}

// ---------------------------------------------------------------------------
// Kernel 3: amplitude embedding (unnormalized; gates are unitary so the
// normalization is folded into the final division by norm^2).
// ---------------------------------------------------------------------------
__global__ void __launch_bounds__(256)
qsim_embed(const float* __restrict__ f, double2* __restrict__ psi) {
    unsigned j = blockIdx.x * blockDim.x + threadIdx.x;
    double2 v; v.x = (double)f[j]; v.y = 0.0;
    psi[j] = v;
}

// ---------------------------------------------------------------------------
// Kernel 4: fused gate group. A block owns BAMPS amplitudes = 2^(EB-r) cosets
// of span(basis); global index j = depBase ^ dep(u) ^ combo(L), LDS index
// e = (L << (EB-r)) | u (u-major => lane-consecutive j => coalesced).
// All gates of the group run in LDS with s_barrier between them.
// ---------------------------------------------------------------------------
__global__ void __launch_bounds__(TPB)
qsim_group(double2* __restrict__ psi,
           const int* __restrict__ gpmask,
           const int* __restrict__ gny,
           const double* __restrict__ cs,
           const QPre* __restrict__ pre,
           int grp) {
    extern __shared__ double2 tile[];                   // BAMPS double2 = 64KB
    if (grp >= pre->ngroups) return;
    const int g0 = pre->gstart[grp], g1 = pre->gend[grp];
    if (g0 >= g1) return;
    const int      r     = pre->grank[grp];
    const int      ub    = EB - r;                      // free (coset-rep) bits per block
    const unsigned umask = (1u << ub) - 1u;

    // deposited base for this block's rep range (low ub bits of t are zero)
    unsigned depBase = 0;
    {
        unsigned tt = blockIdx.x << ub;
        const int nnp = NW - r;
        for (int k = 0; k < nnp; ++k)
            depBase |= ((tt >> k) & 1u) << pre->npbit[grp][k];
    }

    // ---- stage: global -> LDS (async-to-LDS when available) ----
    unsigned jreg[EPT];
    #pragma unroll
    for (int k = 0; k < EPT; ++k) {
        const unsigned e = threadIdx.x + (unsigned)k * TPB;   // [0, BAMPS)
        const unsigned u = e & umask;
        const unsigned L = e >> ub;
        unsigned j = depBase;
        for (int b = 0; b < ub; ++b)
            j ^= ((u >> b) & 1u) << pre->npbit[grp][b];
        for (int i = 0; i < r; ++i)
            if ((L >> i) & 1u) j ^= pre->basis[grp][i];
        jreg[k] = j;
#ifdef QS_ASYNC_LD
        __builtin_amdgcn_global_load_async_to_lds_b128(
            (qs_gptr)(psi + j), (qs_lptr)&tile[e], 0, 0);
#else
        tile[e] = psi[j];
#endif
    }
#ifdef QS_ASYNC_LD
    QS_WAIT_ASYNC();
#endif
    __syncthreads();

    // ---- apply all gates of the group inside LDS ----
    for (int g = g0; g < g1; ++g) {
        const unsigned phi   = pre->phi[g];
        const unsigned pmask = (unsigned)gpmask[g];
        const unsigned upml  = pre->upml[g] & umask;
        const unsigned pmloc = pre->pmloc[g];
        const int      q     = (gny[g] + 1) & 3;
        const double   c     = cs[2 * g + 0];
        const double   s     = cs[2 * g + 1];
        double wr = 0.0, wi = 0.0;                      // w = s * (-i)^q
        if (q == 0)      wr =  s;
        else if (q == 1) wi = -s;
        else if (q == 2) wr = -s;
        else             wi =  s;
        const int bp = __popc(depBase & pmask) & 1;     // block-uniform sign part

        if (phi == 0u) {                                // diagonal (all-Z in-span)
            #pragma unroll
            for (int k = 0; k < EPT; ++k) {
                const unsigned e = threadIdx.x + (unsigned)k * TPB;
                const unsigned u = e & umask, L = e >> ub;
                const int par = (bp + __popc(u & upml) + __popc(L & pmloc)) & 1;
                const double sg = par ? -1.0 : 1.0;
                double2 a = tile[e], rr;
                rr.x = c * a.x + sg * (wr * a.x - wi * a.y);
                rr.y = c * a.y + sg * (wr * a.y + wi * a.x);
                tile[e] = rr;
            }
        } else {
            const unsigned ph  = phi << ub;             // LDS-index flip (u bits are 0)
            const unsigned mb  = 1u << (31 - __clz((int)ph));
            const unsigned lo  = mb - 1u;
            #pragma unroll
            for (int kp = 0; kp < PPT; ++kp) {
                const unsigned p    = threadIdx.x + (unsigned)kp * TPB; // [0, BAMPS/2)
                const unsigned idx0 = ((p & ~lo) << 1) | (p & lo);
                const unsigned idx1 = idx0 ^ ph;
                const unsigned u    = idx0 & umask;     // shared: ph has no u bits
                const unsigned L0   = idx0 >> ub, L1 = idx1 >> ub;
                const int parU = bp + __popc(u & upml);
                const double sg0 = ((parU + __popc(L0 & pmloc)) & 1) ? -1.0 : 1.0;
                const double sg1 = ((parU + __popc(L1 & pmloc)) & 1) ? -1.0 : 1.0;
                double2 a = tile[idx0], b = tile[idx1], ra, rb;
                ra.x = c * a.x + sg0 * (wr * b.x - wi * b.y);
                ra.y = c * a.y + sg0 * (wr * b.y + wi * b.x);
                rb.x = c * b.x + sg1 * (wr * a.x - wi * a.y);
                rb.y = c * b.y + sg1 * (wr * a.y + wi * a.x);
                tile[idx0] = ra;
                tile[idx1] = rb;
            }
        }
        __syncthreads();
    }

    // ---- write back: LDS -> global (async-from-LDS when available) ----
    #pragma unroll
    for (int k = 0; k < EPT; ++k) {
        const unsigned e = threadIdx.x + (unsigned)k * TPB;
#ifdef QS_ASYNC_ST
        __builtin_amdgcn_global_store_async_from_lds_b128(
            (qs_gptr)(psi + jreg[k]), (qs_lptr)&tile[e], 0, 0);
#else
        psi[jreg[k]] = tile[e];
#endif
    }
#ifdef QS_ASYNC_ST
    QS_WAIT_ASYNC();
#endif
}

// ---------------------------------------------------------------------------
// Kernel 5: fused measurement — 8 expvals + norm^2 in one pass (L2-resident
// gathers). Wave32 shuffle reduce -> LDS -> fp64 global atomics.
// ---------------------------------------------------------------------------
__global__ void __launch_bounds__(256)
qsim_meas(const double2* __restrict__ psi,
          const int* __restrict__ mflip,
          const int* __restrict__ mpmask,
          const int* __restrict__ mny,
          double* __restrict__ acc) {
    unsigned fl[NMEAS], pm[NMEAS];
    int      ny[NMEAS];
    #pragma unroll
    for (int m = 0; m < NMEAS; ++m) {
        fl[m] = (unsigned)mflip[m];
        pm[m] = (unsigned)mpmask[m];
        ny[m] = mny[m] & 3;
    }
    double sum[NMEAS + 1];
    #pragma unroll
    for (int m = 0; m <= NMEAS; ++m) sum[m] = 0.0;

    const unsigned stride = gridDim.x * blockDim.x;
    for (unsigned j = blockIdx.x * blockDim.x + threadIdx.x; j < DIM_Q; j += stride) {
        if (j + stride < DIM_Q) __builtin_prefetch(&psi[j + stride], 0, 1);
        double2 a = psi[j];
        sum[NMEAS] += a.x * a.x + a.y * a.y;
        #pragma unroll
        for (int m = 0; m < NMEAS; ++m) {
            double2 b = psi[j ^ fl[m]];
            double dre = a.x * b.x + a.y * b.y;
            double dim = a.x * b.y - a.y * b.x;
            double v = (ny[m] & 1) ? dim : dre;
            if (ny[m] & 2) v = -v;
            double sg = (__popc(j & pm[m]) & 1) ? -1.0 : 1.0;
            sum[m] += sg * v;
        }
    }
    #pragma unroll
    for (int off = 16; off > 0; off >>= 1)
        #pragma unroll
        for (int m = 0; m <= NMEAS; ++m)
            sum[m] += __shfl_down(sum[m], off, 32);

    __shared__ double red[NMEAS + 1][8];
    const int lane = threadIdx.x & 31, wave = threadIdx.x >> 5;
    if (lane == 0)
        #pragma unroll
        for (int m = 0; m <= NMEAS; ++m) red[m][wave] = sum[m];
    __syncthreads();
    if (threadIdx.x == 0) {
        const int nw = blockDim.x >> 5;
        #pragma unroll
        for (int m = 0; m <= NMEAS; ++m) {
            double s = 0.0;
            for (int w = 0; w < nw; ++w) s += red[m][w];
            atomicAdd(&acc[m], s);
        }
    }
}

// ---------------------------------------------------------------------------
// Kernel 6: finalize — deferred normalization.
// ---------------------------------------------------------------------------
__global__ void qsim_final(const double* __restrict__ acc, float* __restrict__ out) {
    int m = threadIdx.x;
    if (m < NMEAS) out[m] = (float)(acc[m] / acc[NMEAS]);
}

// ---------------------------------------------------------------------------
// Host launcher. ws: psi double2[2^21] (32MB) | cs[64],acc[9] | QPre (@+4KB).
// ---------------------------------------------------------------------------
extern "C" void kernel_launch(void* const* d_in, const int* in_sizes, int n_in,
                              void* d_out, int out_size, void* d_ws, size_t ws_size,
                              hipStream_t stream) {
    const float* feature = (const float*)d_in[0];
    const float* theta   = (const float*)d_in[1];
    const int*   gflip   = (const int*)d_in[2];
    const int*   gpmask  = (const int*)d_in[3];
    const int*   gny     = (const int*)d_in[4];
    const int*   mflip   = (const int*)d_in[5];
    const int*   mpmask  = (const int*)d_in[6];
    const int*   mny     = (const int*)d_in[7];

    char*    ws   = (char*)d_ws;
    double2* psi  = (double2*)ws;
    char*    tail = ws + (size_t)DIM_Q * sizeof(double2);
    double*  cs   = (double*)tail;
    double*  acc  = cs + 2 * NGATES;
    QPre*    pre  = (QPre*)(tail + 4096);

    qsim_init<<<1, 64, 0, stream>>>(theta, cs, acc);
    qsim_pre<<<1, 1, 0, stream>>>(gflip, gpmask, pre);
    qsim_embed<<<DIM_Q / 256, 256, 0, stream>>>(feature, psi);

    for (int grp = 0; grp < NGMAX; ++grp)
        qsim_group<<<DIM_Q / BAMPS, TPB, BAMPS * sizeof(double2), stream>>>(
            psi, gpmask, gny, cs, pre, grp);

    qsim_meas<<<1024, 256, 0, stream>>>(psi, mflip, mpmask, mny, acc);
    qsim_final<<<1, 64, 0, stream>>>(acc, (float*)d_out);
}